// Sample_Score_84937273245882
// MI455X (gfx1250) — compile-verified
//
#include <hip/hip_runtime.h>
#include <cstdint>
#include <cstddef>

// Problem constants (fixed by the reference).
#define DD 128
#define SS 2048
#define KK 128
#define GAMMA_F 12.0f
#define PART_STRIDE (SS + 2 * SS * KK)   // p [S] + nt [S*K] + nh [S*K]

// ---------------------------------------------------------------------------
// CDNA5 async global->LDS (ASYNCcnt path). The builtin is declared on this
// toolchain with typed vector pointers:
//   void __builtin_amdgcn_global_load_async_to_lds_b128(
//       global int4* src, local int4* dst, imm int offset, imm int cpol)
// ---------------------------------------------------------------------------
#if defined(__HIP_DEVICE_COMPILE__)

typedef int v4i_ __attribute__((vector_size(16)));
typedef __attribute__((address_space(1))) v4i_* g_v4i_ptr;
typedef __attribute__((address_space(3))) v4i_* l_v4i_ptr;

#if __has_builtin(__builtin_amdgcn_global_load_async_to_lds_b128)
__device__ __forceinline__ void async_ld16(const float* g, float* l) {
  __builtin_amdgcn_global_load_async_to_lds_b128(
      (g_v4i_ptr)(uintptr_t)g,
      (l_v4i_ptr)(uint32_t)(uintptr_t)l,
      0, 0);
}
#else
__device__ __forceinline__ void async_ld16(const float* g, float* l) {
  // VDST = LDS byte address (low 32 bits of the flat shared pointer),
  // VADDR = 64-bit global address, GV mode (saddr = off).
  unsigned lds_off = (unsigned)(uintptr_t)l;
  asm volatile("global_load_async_to_lds_b128 %0, %1, off"
               :: "v"(lds_off), "v"(g) : "memory");
}
#endif

#if __has_builtin(__builtin_amdgcn_s_wait_asynccnt)
#define WAIT_ASYNC(n) __builtin_amdgcn_s_wait_asynccnt(n)
#else
#define WAIT_ASYNC_STR(n) asm volatile("s_wait_asynccnt " #n ::: "memory")
#define WAIT_ASYNC(n) WAIT_ASYNC_STR(n)
#endif

#else  // host pass: stubs so the source parses cleanly
__device__ __forceinline__ void async_ld16(const float*, float*) {}
#define WAIT_ASYNC(n) do {} while (0)
#endif

__device__ __forceinline__ float wave_reduce_add(float v) {
  // wave32 butterfly reduction
  #pragma unroll
  for (int off = 16; off > 0; off >>= 1)
    v += __shfl_xor(v, off, 32);
  return v;
}

// ---------------------------------------------------------------------------
// One block per (part, side, s). 8 wave32s; each wave scores 16 of the K=128
// negatives, staging each 512B embedding row through a private LDS quad
// buffer with async-to-LDS copies pipelined 3 deep.
// ---------------------------------------------------------------------------
__global__ __launch_bounds__(256)
void transe_scores(const float* __restrict__ emb,
                   const float* __restrict__ relG2,
                   const float* __restrict__ relG1,
                   const float* __restrict__ relTO,
                   const int* __restrict__ h1, const int* __restrict__ r1,
                   const int* __restrict__ t1,
                   const int* __restrict__ nt1, const int* __restrict__ nh1,
                   const int* __restrict__ h2, const int* __restrict__ r2,
                   const int* __restrict__ t2,
                   const int* __restrict__ nt2, const int* __restrict__ nh2,
                   const int* __restrict__ h3, const int* __restrict__ t3,
                   const int* __restrict__ nt3, const int* __restrict__ nh3,
                   float* __restrict__ out)
{
  __shared__ __align__(16) float sbuf[8][4][DD];   // 8 waves x 4-deep x 512B

  const int bid  = (int)blockIdx.x;
  const int s    = bid & (SS - 1);
  const int grp  = bid >> 11;     // 0..5
  const int side = grp & 1;       // 0: tail-corrupted (+positive), 1: head
  const int part = grp >> 1;      // 0..2

  int hi, ti;
  const float* rrow;
  const int* negidx;
  size_t obase;
  if (part == 0) {
    hi = h1[s]; ti = t1[s];
    rrow = relG2 + (size_t)r1[s] * DD;
    negidx = side ? nh1 : nt1;
    obase = 0;
  } else if (part == 1) {
    hi = h2[s]; ti = t2[s];
    rrow = relG1 + (size_t)r2[s] * DD;
    negidx = side ? nh2 : nt2;
    obase = (size_t)PART_STRIDE;
  } else {
    hi = h3[s]; ti = t3[s];
    rrow = relTO;                              // broadcast relation
    negidx = side ? nh3 : nt3;
    obase = (size_t)2 * PART_STRIDE;
  }
  negidx += (size_t)s * KK;

  const int tid  = (int)threadIdx.x;
  const int lane = tid & 31;
  const int wave = tid >> 5;

  // Each lane owns dims [4*lane, 4*lane+4).
  const float4 hv = *((const float4*)(emb + (size_t)hi * DD) + lane);
  const float4 tv = *((const float4*)(emb + (size_t)ti * DD) + lane);
  const float4 rv = *((const float4*)rrow + lane);

  float4 q;
  if (side == 0) {            // q = h + r ; score = G - sum|q - row|
    q.x = hv.x + rv.x; q.y = hv.y + rv.y; q.z = hv.z + rv.z; q.w = hv.w + rv.w;
  } else {                    // q = t - r ; |h_neg + r - t| == |h_neg - q|
    q.x = tv.x - rv.x; q.y = tv.y - rv.y; q.z = tv.z - rv.z; q.w = tv.w - rv.w;
  }

  // Positive score: once per s, in the side-0 block, by wave 0.
  if (side == 0 && wave == 0) {
    float p = fabsf(q.x - tv.x) + fabsf(q.y - tv.y) +
              fabsf(q.z - tv.z) + fabsf(q.w - tv.w);
    p = wave_reduce_add(p);
    if (lane == 0) out[obase + (size_t)s] = GAMMA_F - p;
  }

  // Output base for this negative matrix row.
  const size_t odst = obase + SS + (side ? (size_t)SS * KK : 0) + (size_t)s * KK;

  // Wave w handles k = w, w+8, ..., w+120 (16 steps).
#define ISSUE(step) do {                                              \
    const int _k = wave + 8 * (step);                                 \
    const float* _g = emb + (size_t)negidx[_k] * DD + lane * 4;       \
    async_ld16(_g, &sbuf[wave][(step) & 3][lane * 4]);                \
  } while (0)

  ISSUE(0); ISSUE(1); ISSUE(2);

  #pragma unroll
  for (int i = 0; i < 16; ++i) {
    if (i < 13)       { ISSUE(i + 3); WAIT_ASYNC(3); }
    else if (i == 13) { WAIT_ASYNC(2); }
    else if (i == 14) { WAIT_ASYNC(1); }
    else              { WAIT_ASYNC(0); }
    asm volatile("" ::: "memory");   // order LDS reads after the async wait

    const float4 nv = *((const float4*)&sbuf[wave][i & 3][0] + lane);
    float p = fabsf(q.x - nv.x) + fabsf(q.y - nv.y) +
              fabsf(q.z - nv.z) + fabsf(q.w - nv.w);
    p = wave_reduce_add(p);
    if (lane == 0) out[odst + (size_t)(wave + 8 * i)] = GAMMA_F - p;
  }
#undef ISSUE
}

extern "C" void kernel_launch(void* const* d_in, const int* in_sizes, int n_in,
                              void* d_out, int out_size, void* d_ws, size_t ws_size,
                              hipStream_t stream) {
  (void)in_sizes; (void)n_in; (void)out_size; (void)d_ws; (void)ws_size;
  const float* emb = (const float*)d_in[0];
  const float* rG2 = (const float*)d_in[1];
  const float* rG1 = (const float*)d_in[2];
  const float* rTO = (const float*)d_in[3];
  const int* h1  = (const int*)d_in[4];
  const int* r1  = (const int*)d_in[5];
  const int* t1  = (const int*)d_in[6];
  const int* nt1 = (const int*)d_in[7];
  const int* nh1 = (const int*)d_in[8];
  const int* h2  = (const int*)d_in[9];
  const int* r2  = (const int*)d_in[10];
  const int* t2  = (const int*)d_in[11];
  const int* nt2 = (const int*)d_in[12];
  const int* nh2 = (const int*)d_in[13];
  const int* h3  = (const int*)d_in[14];
  const int* t3  = (const int*)d_in[15];
  const int* nt3 = (const int*)d_in[16];
  const int* nh3 = (const int*)d_in[17];
  float* out = (float*)d_out;

  dim3 grid(6 * SS);   // 3 parts x 2 sides x S
  dim3 block(256);
  transe_scores<<<grid, block, 0, stream>>>(emb, rG2, rG1, rTO,
                                            h1, r1, t1, nt1, nh1,
                                            h2, r2, t2, nt2, nh2,
                                            h3, t3, nt3, nh3,
                                            out);
}